// LinkPredictor_82952998355939
// MI455X (gfx1250) — compile-verified
//
#include <hip/hip_runtime.h>
#include <hip/hip_bf16.h>

// ---------------------------------------------------------------------------
// 2-layer GCN for MI455X (gfx1250, wave32).
//   h1 = A_norm @ (x @ W1) + b1 ; relu ; out = A_norm @ (h1 @ W2) + b2
// Dense GEMMs: fp32 WMMA pipe (V_WMMA_F32_16X16X4_F32), full fp32 precision.
// Sparse aggregation: L2-resident (192MB L2 >> all buffers) float4-vectorized
// gather + 4x global_atomic_add_f32 per thread.
// ---------------------------------------------------------------------------

typedef __attribute__((ext_vector_type(2))) float v2f;
typedef __attribute__((ext_vector_type(8))) float v8f;

// ---- degree / normalization ------------------------------------------------

__global__ void init_deg_kernel(float* __restrict__ deg, int N) {
    int i = blockIdx.x * blockDim.x + threadIdx.x;
    if (i < N) deg[i] = 1.0f;  // self-loop contributes 1
}

__global__ void deg_count_kernel(const long long* __restrict__ dst,
                                 float* __restrict__ deg, int E) {
    int e = blockIdx.x * blockDim.x + threadIdx.x;
    if (e < E) atomicAdd(&deg[(int)dst[e]], 1.0f);
}

__global__ void dinv_kernel(float* __restrict__ deg, int N) {
    int i = blockIdx.x * blockDim.x + threadIdx.x;
    if (i < N) deg[i] = rsqrtf(deg[i]);
}

// ---- fp32 WMMA GEMM: C[N,NCOL] = A[N,K] @ W[K,NCOL] ------------------------
// One wave computes one 16x16 output tile; wave w in the block owns column
// tile w. Requires N % 16 == 0 (holds: 50000 = 3125*16), K % 4 == 0.
//
// V_WMMA_F32_16X16X4_F32 operand layout (ISA 7.12.2):
//   A 16x4 : lanes 0-15 -> M=lane, VGPR{0,1}=K{0,1}; lanes 16-31 -> K{2,3}
//   B 4x16 : lanes 0-15 -> N=lane, VGPR{0,1}=K{0,1}; lanes 16-31 -> K{2,3}
//   C/D    : VGPR v, lane l -> D[v + 8*(l/16)][l%16]

template <int K, int NCOL>
__global__ void gemm_wmma_f32(const float* __restrict__ A,
                              const float* __restrict__ W,
                              float* __restrict__ C) {
    const int lane = threadIdx.x & 31;
    const int wave = threadIdx.x >> 5;
    const int row0 = blockIdx.x << 4;   // 16 rows per block
    const int col0 = wave << 4;         // 16 cols per wave
    const int grp  = lane >> 4;         // 0: K pair {0,1}, 1: K pair {2,3}
    const int lr   = lane & 15;

    const float* arow = A + (size_t)(row0 + lr) * K;  // this lane's A row
    const float* wcol = W + (col0 + lr);              // this lane's B column

    v8f acc = {};
#pragma unroll
    for (int k = 0; k < K; k += 4) {
        const int ka = k + grp * 2;
        v2f a = *(const v2f*)(arow + ka);     // two consecutive K of one row
        v2f b;
        b.x = wcol[(size_t)ka * NCOL];        // W[ka  ][col]
        b.y = wcol[(size_t)(ka + 1) * NCOL];  // W[ka+1][col]
        acc = __builtin_amdgcn_wmma_f32_16x16x4_f32(
            /*neg_a=*/false, a, /*neg_b=*/false, b,
            /*c_mod=*/(short)0, acc, /*reuse_a=*/false, /*reuse_b=*/false);
    }

    float* crow = C + (size_t)(row0 + grp * 8) * NCOL + col0 + lr;
#pragma unroll
    for (int v = 0; v < 8; ++v)
        crow[(size_t)v * NCOL] = acc[v];
}

// ---- self-loop init (non-atomic, float4): out = h * dinv^2 (+ bias) --------
// One thread per 4 channels. LOGC = log2(channels), channels % 4 == 0.

template <int LOGC, bool BIAS>
__global__ void selfloop4_kernel(const float* __restrict__ h,
                                 const float* __restrict__ dinv,
                                 const float* __restrict__ bias,
                                 float* __restrict__ out, int total4) {
    constexpr int LOGCV = LOGC - 2;            // float4 groups per node
    int i = blockIdx.x * blockDim.x + threadIdx.x;
    if (i >= total4) return;
    const float dv = dinv[i >> LOGCV];
    const float s  = dv * dv;
    float4 hv = *(const float4*)(h + ((size_t)i << 2));
    float4 r;
    r.x = hv.x * s; r.y = hv.y * s; r.z = hv.z * s; r.w = hv.w * s;
    if (BIAS) {
        const float4 bv = *(const float4*)(bias + ((i & ((1 << LOGCV) - 1)) << 2));
        r.x += bv.x; r.y += bv.y; r.z += bv.z; r.w += bv.w;
    }
    *(float4*)(out + ((size_t)i << 2)) = r;
}

// ---- edge scatter (float4): out[dst] += h[src] * dinv[src]*dinv[dst] -------
// One thread per (edge, 4 channels): indices + norm loaded once, one b128
// gather, four back-to-back global_atomic_add_f32 for MLP.

template <int LOGC>
__global__ void scatter4_kernel(const float* __restrict__ h,
                                const float* __restrict__ dinv,
                                const long long* __restrict__ src,
                                const long long* __restrict__ dst,
                                float* __restrict__ out, int total4) {
    constexpr int LOGCV = LOGC - 2;            // float4 groups per node
    int idx = blockIdx.x * blockDim.x + threadIdx.x;
    if (idx >= total4) return;
    const int e  = idx >> LOGCV;
    const int c4 = (idx & ((1 << LOGCV) - 1)) << 2;
    const int s  = (int)src[e];
    const int d  = (int)dst[e];
    const float nrm = dinv[s] * dinv[d];
    const float4 hv = *(const float4*)(h + ((size_t)s << LOGC) + c4);
    float* o = out + ((size_t)d << LOGC) + c4;
    atomicAdd(o + 0, hv.x * nrm);
    atomicAdd(o + 1, hv.y * nrm);
    atomicAdd(o + 2, hv.z * nrm);
    atomicAdd(o + 3, hv.w * nrm);
}

// ---- bias + ReLU (float4) --------------------------------------------------

template <int LOGC>
__global__ void bias_relu4_kernel(const float* __restrict__ in,
                                  const float* __restrict__ bias,
                                  float* __restrict__ out, int total4) {
    constexpr int LOGCV = LOGC - 2;
    int i = blockIdx.x * blockDim.x + threadIdx.x;
    if (i >= total4) return;
    float4 v = *(const float4*)(in + ((size_t)i << 2));
    const float4 bv = *(const float4*)(bias + ((i & ((1 << LOGCV) - 1)) << 2));
    v.x = fmaxf(v.x + bv.x, 0.0f);
    v.y = fmaxf(v.y + bv.y, 0.0f);
    v.z = fmaxf(v.z + bv.z, 0.0f);
    v.w = fmaxf(v.w + bv.w, 0.0f);
    *(float4*)(out + ((size_t)i << 2)) = v;
}

// ---------------------------------------------------------------------------

extern "C" void kernel_launch(void* const* d_in, const int* in_sizes, int n_in,
                              void* d_out, int out_size, void* d_ws, size_t ws_size,
                              hipStream_t stream) {
    const float*     x   = (const float*)d_in[0];      // [N,128] f32
    const long long* ei  = (const long long*)d_in[1];  // [2,E] int64
    const float*     W1  = (const float*)d_in[2];      // [128,64]
    const float*     b1  = (const float*)d_in[3];      // [64]
    const float*     W2  = (const float*)d_in[4];      // [64,32]
    const float*     b2  = (const float*)d_in[5];      // [32]
    float*           out = (float*)d_out;              // [N,32]

    const int N = in_sizes[0] / 128;
    const int E = in_sizes[1] / 2;
    const long long* srcp = ei;
    const long long* dstp = ei + E;

    // Workspace layout (floats): dinv | h1[N,64] | hagg[N,64] | h2[N,32]
    float* ws   = (float*)d_ws;
    float* dinv = ws;
    float* h1   = ws + (((size_t)N + 63) & ~(size_t)63);
    float* hagg = h1 + (size_t)N * 64;
    float* h2   = hagg + (size_t)N * 64;

    const int T = 256;
    const int gN = (N + T - 1) / T;
    const int gE = (E + T - 1) / T;

    const int n64_4 = N * (64 / 4);            // float4 groups, C=64
    const int n32_4 = N * (32 / 4);            // float4 groups, C=32
    const int e64_4 = (int)((size_t)E * (64 / 4));
    const int e32_4 = (int)((size_t)E * (32 / 4));

    // Normalization: deg = 1 + in-degree; dinv = rsqrt(deg)
    init_deg_kernel<<<gN, T, 0, stream>>>(dinv, N);
    deg_count_kernel<<<gE, T, 0, stream>>>(dstp, dinv, E);
    dinv_kernel<<<gN, T, 0, stream>>>(dinv, N);

    // Layer 1: h1 = x @ W1 (WMMA fp32), aggregate, bias+relu
    gemm_wmma_f32<128, 64><<<N / 16, 128, 0, stream>>>(x, W1, h1);
    selfloop4_kernel<6, false><<<(n64_4 + T - 1) / T, T, 0, stream>>>(h1, dinv, nullptr, hagg, n64_4);
    scatter4_kernel<6><<<(e64_4 + T - 1) / T, T, 0, stream>>>(h1, dinv, srcp, dstp, hagg, e64_4);
    bias_relu4_kernel<6><<<(n64_4 + T - 1) / T, T, 0, stream>>>(hagg, b1, h1, n64_4);

    // Layer 2: h2 = relu(h1) @ W2 (WMMA fp32), aggregate + bias into d_out
    gemm_wmma_f32<64, 32><<<N / 16, 64, 0, stream>>>(h1, W2, h2);
    selfloop4_kernel<5, true><<<(n32_4 + T - 1) / T, T, 0, stream>>>(h2, dinv, b2, out, n32_4);
    scatter4_kernel<5><<<(e32_4 + T - 1) / T, T, 0, stream>>>(h2, dinv, srcp, dstp, out, e32_4);
}